// RangeIndexedLinear_34256659153070
// MI455X (gfx1250) — compile-verified
//
#include <hip/hip_runtime.h>
#include <hip/hip_bf16.h>
#include <stdint.h>

// ---------------------------------------------------------------------------
// CDNA5 wave32 reduction via V_WMMA_F32_16X16X4_F32.
//
// A (16x4 f32, 2 VGPRs): a.x = per-lane partial -> A[m][0]=acc(lane m),
//   A[m][2]=acc(lane m+16); a.y = 0 -> A[m][1]=A[m][3]=0.
// B (4x16 f32, 2 VGPRs): all 1.0 -> D[m][n] = rowsum(A row m) = acc(m)+acc(m+16).
// D (16x16 f32, 8 VGPRs): lanes 0-15 hold rows 0-7, lanes 16-31 hold rows 8-15.
// Per-lane sum of the 8 D VGPRs = half-wave sum; shfl_xor(16) completes it.
// Exact f32 arithmetic (RNE), so numerically equivalent to a shuffle tree.
// ---------------------------------------------------------------------------
typedef float v2f __attribute__((ext_vector_type(2)));
typedef float v8f __attribute__((ext_vector_type(8)));

__device__ __forceinline__ float wave_reduce_sum_wmma(float acc) {
  v2f a; a.x = acc;  a.y = 0.0f;
  v2f b; b.x = 1.0f; b.y = 1.0f;
  v8f c = {};
  // 8 args: (neg_a, A, neg_b, B, c_mod, C, reuse_a, reuse_b)
  c = __builtin_amdgcn_wmma_f32_16x16x4_f32(false, a, false, b, (short)0, c,
                                            false, false);
  float t = c[0] + c[1] + c[2] + c[3] + c[4] + c[5] + c[6] + c[7];
  t += __shfl_xor(t, 16, 32);   // combine the two half-wave sums
  return t;                     // full 32-lane sum, uniform across the wave
}

__device__ __forceinline__ float block_reduce_sum(float acc, float* s_wave) {
  const int lane = threadIdx.x & 31;
  const int wid  = threadIdx.x >> 5;
  float w = wave_reduce_sum_wmma(acc);
  if (lane == 0) s_wave[wid] = w;
  __syncthreads();
  float bs = 0.0f;
  if (threadIdx.x == 0) {
    const int nw = (blockDim.x + 31) >> 5;
    for (int i = 0; i < nw; ++i) bs += s_wave[i];  // fixed order: deterministic
  }
  return bs;  // valid in thread 0 only
}

// ---------------------------------------------------------------------------
// Kernel 1: v_i = mean_b x[b][i]; bucket search in LDS tables; gated gather of
// weights; per-block partial dot product.
// ---------------------------------------------------------------------------
__global__ void __launch_bounds__(256)
rangedot_partial_kernel(const float* __restrict__ x,
                        const float* __restrict__ weights,
                        const float* __restrict__ min_vals,
                        const float* __restrict__ max_vals,
                        const int*   __restrict__ start_pos,
                        const int*   __restrict__ offsets,
                        const int*   __restrict__ sizes,
                        int B, int D, int G, long long wlen,
                        float* __restrict__ partials) {
  extern __shared__ unsigned char smem_raw[];
  float* s_min   = reinterpret_cast<float*>(smem_raw);
  float* s_max   = s_min + G;
  int*   s_start = reinterpret_cast<int*>(s_max + G);
  int*   s_off   = s_start + G;
  int*   s_size  = s_off + G;
  __shared__ float s_wave[32];

  // Stage the 5 lookup tables (5*G*4 B = 20 KB for G=1024) into LDS so the
  // per-lane divergent binary search + table reads never touch HBM.
  for (int t = threadIdx.x; t < G; t += blockDim.x) {
    s_min[t]   = min_vals[t];
    s_max[t]   = max_vals[t];
    s_start[t] = start_pos[t];
    s_off[t]   = offsets[t];
    s_size[t]  = sizes[t];
  }
  __syncthreads();

  const float invB = 1.0f / (float)B;
  float acc = 0.0f;

  for (int i = blockIdx.x * blockDim.x + threadIdx.x; i < D;
       i += gridDim.x * blockDim.x) {
    // v = mean over B rows (coalesced: consecutive lanes -> consecutive i)
    float vs = 0.0f;
    for (int bb = 0; bb < B; ++bb) vs += x[(size_t)bb * (size_t)D + i];
    const float v = vs * invB;

    // g = searchsorted(min_vals, v, 'right') - 1  (upper_bound - 1)
    int lo = 0, hi = G;
    while (lo < hi) {
      int mid = (lo + hi) >> 1;
      if (s_min[mid] <= v) lo = mid + 1; else hi = mid;
    }
    const int  g  = lo - 1;
    int gc = g; if (gc < 0) gc = 0; if (gc > G - 1) gc = G - 1;

    const bool in_range = (g >= 0) && (v >= s_min[gc]) && (v <= s_max[gc]);
    const int  pos      = i - s_start[gc];
    const bool valid    = in_range && (pos >= 0) && (pos < s_size[gc]);

    long long widx = (long long)s_off[gc] + (long long)pos;
    if (widx < 0) widx = 0;
    if (widx > wlen - 1) widx = wlen - 1;

    const float w = valid ? weights[widx] : 0.0f;
    acc += v * w;
  }

  // All lanes converged here; EXEC is all-ones for the WMMA.
  const float bs = block_reduce_sum(acc, s_wave);
  if (threadIdx.x == 0) partials[blockIdx.x] = bs;
}

// ---------------------------------------------------------------------------
// Kernel 2: single-block deterministic reduce of block partials -> scalar s.
// ---------------------------------------------------------------------------
__global__ void __launch_bounds__(256)
rangedot_final_kernel(const float* __restrict__ partials, int n,
                      float* __restrict__ s_out) {
  __shared__ float s_wave[32];
  float acc = 0.0f;
  for (int i = threadIdx.x; i < n; i += blockDim.x) acc += partials[i];
  const float bs = block_reduce_sum(acc, s_wave);
  if (threadIdx.x == 0) *s_out = bs;
}

// ---------------------------------------------------------------------------
// Kernel 3: out[0][j] = mask[j] ? s : 0 ; rows 1..B-1 = 0.
// ---------------------------------------------------------------------------
__global__ void __launch_bounds__(256)
rangedot_write_kernel(const float* __restrict__ s_ptr,
                      const uint8_t* __restrict__ mask,
                      float* __restrict__ out, int brows, int outw) {
  const int idx = blockIdx.x * blockDim.x + threadIdx.x;
  const int total = brows * outw;
  if (idx >= total) return;
  const int r = idx / outw;
  const int j = idx - r * outw;
  const float s = *s_ptr;  // broadcast scalar (L2-resident)
  out[idx] = (r == 0 && mask[j] != 0) ? s : 0.0f;
}

// ---------------------------------------------------------------------------
// Host side
// ---------------------------------------------------------------------------
extern "C" void kernel_launch(void* const* d_in, const int* in_sizes, int n_in,
                              void* d_out, int out_size, void* d_ws,
                              size_t ws_size, hipStream_t stream) {
  (void)n_in; (void)ws_size;
  const float* x         = (const float*)d_in[0];
  const float* weights   = (const float*)d_in[1];
  const float* min_vals  = (const float*)d_in[2];
  const float* max_vals  = (const float*)d_in[3];
  const int*   start_pos = (const int*)d_in[4];
  const int*   offsets   = (const int*)d_in[5];
  const int*   sizes     = (const int*)d_in[6];
  const uint8_t* mask    = (const uint8_t*)d_in[7];
  float* out = (float*)d_out;

  const int G = in_sizes[2];                       // number of ranges
  const int D = in_sizes[1] / G;                   // weights = G*D
  const int B = in_sizes[0] / D;                   // x = B*D
  const long long wlen = (long long)in_sizes[1];
  const int OUTW = in_sizes[7];
  const int BROWS = out_size / OUTW;

  int nblocks = (D + 255) / 256;
  if (nblocks > 512) nblocks = 512;                // grid-stride covers the rest
  if (nblocks < 1) nblocks = 1;

  float* partials = (float*)d_ws;                  // [nblocks] partials
  float* s_scalar = partials + nblocks;            // [1] final dot product

  const size_t lds_bytes = (size_t)G * 4u * 5u;    // 20 KB for G=1024

  rangedot_partial_kernel<<<nblocks, 256, lds_bytes, stream>>>(
      x, weights, min_vals, max_vals, start_pos, offsets, sizes,
      B, D, G, wlen, partials);

  rangedot_final_kernel<<<1, 256, 0, stream>>>(partials, nblocks, s_scalar);

  const int total = BROWS * OUTW;
  rangedot_write_kernel<<<(total + 255) / 256, 256, 0, stream>>>(
      s_scalar, mask, out, BROWS, OUTW);
}